// GraphSAGE_59133109732147
// MI455X (gfx1250) — compile-verified
//
#include <hip/hip_runtime.h>

typedef __attribute__((ext_vector_type(2))) float v2f;
typedef __attribute__((ext_vector_type(8))) float v8f;
typedef int v4i __attribute__((vector_size(16)));

#define N_NODES 100000
#define KNB 16
#define DH 128
#define NODES_PER_BLK 16
#define THREADS 512
#define WA_STRIDE 132   // 132 % 64 == 4 -> conflict-free b64 B-tile reads
#define AGG_STRIDE 132
#define OUT_STRIDE 132

#if __has_builtin(__builtin_amdgcn_global_load_async_to_lds_b128)
#define HAVE_ASYNC_LDS 1
#else
#define HAVE_ASYNC_LDS 0
#endif

typedef __attribute__((address_space(1))) v4i gv4i;   // global 16B vector
typedef __attribute__((address_space(3))) v4i lv4i;   // LDS 16B vector

// One-time transpose of the weight tensors into the workspace:
//   WaT[l][h][d] = Wa[l][d][h]   (128x128 per layer)
//   WlT[l][h][k] = Wl[l][k][h]   (256x128 per layer -> 128x256 transposed)
__global__ void transpose_weights_kernel(const float* __restrict__ Wa,
                                         const float* __restrict__ Wl,
                                         float* __restrict__ WaT,
                                         float* __restrict__ WlT)
{
    const int i = blockIdx.x * blockDim.x + threadIdx.x;
    if (i < 3 * DH * DH) {
        const int l = i / (DH * DH), r = i % (DH * DH);
        const int d = r >> 7, h = r & 127;
        WaT[l * DH * DH + h * DH + d] = Wa[i];
    }
    if (i < 3 * 2 * DH * DH) {
        const int l = i / (2 * DH * DH), r = i % (2 * DH * DH);
        const int k = r >> 7, h = r & 127;
        WlT[l * 2 * DH * DH + h * 2 * DH + k] = Wl[i];
    }
}

// Fused GraphSAGE layer:
//  phase 1: per-node neighbor GEMM  relu(NB[16,128] @ Wa) -> max-pool -> agg[128]   (wave per node)
//  phase 2: concat(self, agg)[16,256] @ Wl[256,128] + bl                            (waves 0..7)
//  phase 3: LayerNorm + ReLU + residual (or plain writeout for the last layer)
__launch_bounds__(THREADS)
__global__ void sage_layer_kernel(
    const float* __restrict__ x_in,   // [N,128]
    float*       __restrict__ x_out,  // [N,128]
    const int*   __restrict__ nidx,   // [N,16]
    const float* __restrict__ WaT,    // [128,128]  (h,d)  transposed aggregator weight
    const float* __restrict__ ba,     // [128]
    const float* __restrict__ WlT,    // [128,256]  (h,k)  transposed concat weight
    const float* __restrict__ bl,     // [128]
    const float* __restrict__ lng,    // [128]
    const float* __restrict__ lnb,    // [128]
    int apply_ln)
{
    __shared__ float sWaT[DH * WA_STRIDE];            // [h][d], padded stride
    __shared__ float sAgg[NODES_PER_BLK * AGG_STRIDE];
    __shared__ float sOut[NODES_PER_BLK * OUT_STRIDE];
    __shared__ int   sIdx[NODES_PER_BLK * KNB];

    const int tid  = threadIdx.x;
    const int wave = tid >> 5;
    const int lane = tid & 31;
    const int base = blockIdx.x * NODES_PER_BLK;      // N divisible by 16

    // ---- stage WaT and neighbor indices into LDS ----
#if HAVE_ASYNC_LDS
    for (int v = tid; v < DH * DH / 4; v += THREADS) {
        const int idx = v * 4;                        // 4 consecutive floats, same h row
        const int h = idx >> 7, d = idx & 127;
        gv4i* gsrc = (gv4i*)(unsigned long long)(const float*)&WaT[idx];
        lv4i* ldst = (lv4i*)(unsigned int)(unsigned long long)&sWaT[h * WA_STRIDE + d];
        __builtin_amdgcn_global_load_async_to_lds_b128(gsrc, ldst, 0, 0);
    }
#if __has_builtin(__builtin_amdgcn_s_wait_asynccnt)
    __builtin_amdgcn_s_wait_asynccnt(0);
#else
    asm volatile("s_wait_asynccnt 0" ::: "memory");
#endif
#else
    for (int idx = tid; idx < DH * DH; idx += THREADS) {
        const int h = idx >> 7, d = idx & 127;
        sWaT[h * WA_STRIDE + d] = WaT[idx];           // coalesced, conflict-free
    }
#endif
    if (tid < NODES_PER_BLK * KNB) sIdx[tid] = nidx[base * KNB + tid];
    __syncthreads();

    // ---------------- phase 1: aggregator (one wave per node) ----------------
    {
        const int node_local = wave;
        const int r   = lane & 15;                    // neighbor row (A-matrix M)
        const int kh  = lane >> 4;                    // K-half: lanes 16..31 hold K+2,K+3
        const int col = lane & 15;                    // B/D column within tile
        const long nbr = (long)sIdx[node_local * KNB + r];
        const float* arow = x_in + nbr * DH + 2 * kh;

        v8f acc[8];
        const v8f zero8 = {};
        #pragma unroll
        for (int t = 0; t < 8; ++t) acc[t] = zero8;

        for (int k0 = 0; k0 < DH; k0 += 4) {
            const v2f a = *(const v2f*)(arow + k0);   // A 16x4 slice, layout per ISA 7.12.2
            #pragma unroll
            for (int t = 0; t < 8; ++t) {
                const v2f b = *(const v2f*)(&sWaT[(t * 16 + col) * WA_STRIDE + k0 + 2 * kh]);
                acc[t] = __builtin_amdgcn_wmma_f32_16x16x4_f32(
                    false, a, false, b, (short)0, acc[t], false, false);
            }
        }
        // max over 16 neighbor rows; relu(max+bias) == max(relu(row+bias)) since bias is per-col
        #pragma unroll
        for (int t = 0; t < 8; ++t) {
            float m = acc[t][0];
            #pragma unroll
            for (int rr = 1; rr < 8; ++rr) m = fmaxf(m, acc[t][rr]);
            m = fmaxf(m, __shfl_xor(m, 16, 32));      // fold M and M+8 half-waves
            const float val = fmaxf(m + ba[t * 16 + col], 0.0f);
            if (lane < 16) sAgg[node_local * AGG_STRIDE + t * 16 + col] = val;
        }
    }
    __syncthreads();

    // ---------------- phase 2: concat-linear (waves 0..7, N-tile = wave) ----------------
    if (wave < 8) {
        const int m   = lane & 15;                    // node row
        const int kh  = lane >> 4;
        const int col = lane & 15;
        const float* xrow = x_in + (long)(base + m) * DH + 2 * kh;
        const float* wrow = WlT + (wave * 16 + col) * 2 * DH + 2 * kh;   // [k] contiguous

        v8f acc = {};
        for (int k0 = 0; k0 < DH; k0 += 4) {          // self-feature half (k < 128)
            const v2f a = *(const v2f*)(xrow + k0);
            const v2f b = *(const v2f*)(wrow + k0);
            acc = __builtin_amdgcn_wmma_f32_16x16x4_f32(
                false, a, false, b, (short)0, acc, false, false);
        }
        for (int k0 = 0; k0 < DH; k0 += 4) {          // aggregated half (k >= 128)
            const v2f a = *(const v2f*)(&sAgg[m * AGG_STRIDE + k0 + 2 * kh]);
            const v2f b = *(const v2f*)(wrow + DH + k0);
            acc = __builtin_amdgcn_wmma_f32_16x16x4_f32(
                false, a, false, b, (short)0, acc, false, false);
        }
        const float bias = bl[wave * 16 + col];
        #pragma unroll
        for (int rr = 0; rr < 8; ++rr) {              // D: VGPR rr -> M = rr + 8*kh
            sOut[(rr + 8 * kh) * OUT_STRIDE + wave * 16 + col] = acc[rr] + bias;
        }
    }
    __syncthreads();

    // ---------------- phase 3: LayerNorm + ReLU + residual, or writeout ----------------
    if (apply_ln) {
        float v[4];
        float s = 0.0f, s2 = 0.0f;                    // wave w handles node w
        #pragma unroll
        for (int j = 0; j < 4; ++j) {
            v[j] = sOut[wave * OUT_STRIDE + lane + 32 * j];
            s  += v[j];
            s2 += v[j] * v[j];
        }
        #pragma unroll
        for (int off = 16; off >= 1; off >>= 1) {
            s  += __shfl_xor(s,  off, 32);
            s2 += __shfl_xor(s2, off, 32);
        }
        const float mu   = s * (1.0f / DH);
        const float var  = s2 * (1.0f / DH) - mu * mu;   // biased variance (jnp.var)
        const float rstd = __frsqrt_rn(var + 1e-5f);
        float* orow = x_out + (long)(base + wave) * DH;
        #pragma unroll
        for (int j = 0; j < 4; ++j) {
            const int c = lane + 32 * j;
            const float xn = (v[j] - mu) * rstd * lng[c] + lnb[c];
            orow[c] = fmaxf(xn, 0.0f) + v[j];
        }
    } else {
        #pragma unroll
        for (int j = 0; j < 4; ++j) {
            const int idx = tid + THREADS * j;        // 0..2047, coalesced
            const int m = idx >> 7, c = idx & 127;
            x_out[(long)(base + m) * DH + c] = sOut[m * OUT_STRIDE + c];
        }
    }
}

extern "C" void kernel_launch(void* const* d_in, const int* in_sizes, int n_in,
                              void* d_out, int out_size, void* d_ws, size_t ws_size,
                              hipStream_t stream) {
    (void)in_sizes; (void)n_in; (void)out_size; (void)ws_size;
    const float* features = (const float*)d_in[0];   // [N,128]
    const int*   nidx     = (const int*)d_in[1];     // [N,16]
    const float* Wl       = (const float*)d_in[2];   // [3,256,128]
    const float* bl       = (const float*)d_in[3];   // [3,128]
    const float* lng      = (const float*)d_in[4];   // [2,128]
    const float* lnb      = (const float*)d_in[5];   // [2,128]
    const float* Wa       = (const float*)d_in[6];   // [3,128,128]
    const float* ba       = (const float*)d_in[7];   // [3,128]

    float* x0  = (float*)d_ws;                                // ping  [N,128]
    float* x1  = x0 + (size_t)N_NODES * DH;                   // pong  [N,128]
    float* WaT = x1 + (size_t)N_NODES * DH;                   // [3,128,128]
    float* WlT = WaT + (size_t)3 * DH * DH;                   // [3,128,256]
    float* out = (float*)d_out;

    // one-time weight transposes (deterministic, every call)
    transpose_weights_kernel<<<dim3((3 * 2 * DH * DH + 255) / 256), dim3(256), 0, stream>>>(
        Wa, Wl, WaT, WlT);

    const dim3 grid(N_NODES / NODES_PER_BLK);
    const dim3 block(THREADS);

    // layer 0: features -> x0 (LN)
    sage_layer_kernel<<<grid, block, 0, stream>>>(features, x0, nidx,
        WaT, ba, WlT, bl, lng, lnb, 1);
    // layer 1: x0 -> x1 (LN)
    sage_layer_kernel<<<grid, block, 0, stream>>>(x0, x1, nidx,
        WaT + DH * DH, ba + DH, WlT + 2 * DH * DH, bl + DH, lng + DH, lnb + DH, 1);
    // layer 2: x1 -> out (no LN)
    sage_layer_kernel<<<grid, block, 0, stream>>>(x1, out, nidx,
        WaT + 2 * DH * DH, ba + 2 * DH, WlT + 4 * DH * DH, bl + 2 * DH, lng, lnb, 0);
}